// DrQv2Aug_36696200577237
// MI455X (gfx1250) — compile-verified
//
#include <hip/hip_runtime.h>

// DrQv2 random-shift augmentation for MI455X (gfx1250).
//
// Pure data movement (no FLOPs): 130 MB in + 130 MB out -> ~11 us floor at
// 23.3 TB/s. Strategy: per (image, channel) plane, use the CDNA5 Tensor Data
// Mover to DMA the whole 84x84 f32 plane (28224 B) global -> LDS in a single
// tensor_load_to_lds, then do the replicate-clamped shifted gather out of LDS
// and write coalesced, 16B-aligned non-temporal b128 stores.

namespace {
constexpr int PADS  = 4;
constexpr int H     = 84;
constexpr int W     = 84;
constexpr int C     = 9;
constexpr int PLANE = H * W;      // 7056 elements = 28224 bytes
constexpr int ROWQ  = W / 4;      // 21 float4 per row
constexpr int QUADS = PLANE / 4;  // 1764 float4 per plane

typedef __attribute__((ext_vector_type(4))) unsigned int v4u;
typedef __attribute__((ext_vector_type(8))) int          v8i;
typedef __attribute__((ext_vector_type(4))) int          v4i;
typedef __attribute__((ext_vector_type(4))) float        v4f;
} // namespace

__global__ __launch_bounds__(256) void drqv2_shift_tdm(
    const float* __restrict__ x,
    const int*   __restrict__ shift,
    float*       __restrict__ out)
{
    __shared__ float plane[PLANE];

    const int ch  = blockIdx.x;   // 0..8
    const int img = blockIdx.y;   // 0..B*L-1
    const long long plane_off = ((long long)img * C + ch) * (long long)PLANE;

    // ---- TDM: DMA this plane global -> LDS (one descriptor, wave 0 only) ----
    if (threadIdx.x < 32) {
        const unsigned long long gaddr =
            (unsigned long long)(const void*)(x + plane_off);
        // Flat LDS address low 32 bits == LDS byte offset (aperture rule).
        const unsigned int lds_addr =
            (unsigned int)(unsigned long long)(void*)&plane[0];

        // D# group 0: count=1 (valid user descriptor), lds_addr,
        // 57-bit global_addr, type=2 ("image") in bits [127:126].
        v4u g0;
        g0.x = 1u;
        g0.y = lds_addr;
        g0.z = (unsigned int)(gaddr & 0xFFFFFFFFull);
        g0.w = (unsigned int)((gaddr >> 32) & 0x01FFFFFFull) | (2u << 30);

        // D# group 1: no multicast, data_size=2 (4 bytes), no barrier/pad,
        // 1-D tensor: tensor_dim0 = tile_dim0 = 7056, tensor_dim1 = 1,
        // tile_dim1 = tile_dim2 = 0 (unused), tensor_dim0_stride = 7056.
        v8i g1;
        g1[0] = (int)(2u << 16);                              // data_size = 4B
        g1[1] = (int)(((unsigned)PLANE & 0xFFFFu) << 16);     // dim0 lo16
        g1[2] = (int)((((unsigned)PLANE >> 16) & 0xFFFFu)     // dim0 hi16
                      | (1u << 16));                          // dim1 lo16 = 1
        g1[3] = (int)(((unsigned)PLANE & 0xFFFFu) << 16);     // tile_dim0
        g1[4] = 0;                                            // tile_dim1/2 = 0
        g1[5] = PLANE;                                        // dim0 stride
        g1[6] = 0;
        g1[7] = 0;

        v4i z4 = {0, 0, 0, 0};          // groups 2/3 unused (<=2D tensor)
        v8i z8 = {0, 0, 0, 0, 0, 0, 0, 0};
        __builtin_amdgcn_tensor_load_to_lds(g0, g1, z4, z4, z8, 0);
        __builtin_amdgcn_s_wait_tensorcnt(0);   // DMA complete for this wave
    }
    __syncthreads();                            // publish LDS to all 8 waves

    // Per-image integer shift: col 0 = x-shift, col 1 = y-shift (uniform/block).
    const int dx = shift[2 * img + 0] - PADS;   // in [-4, 4]
    const int dy = shift[2 * img + 1] - PADS;

    float* outp = out + plane_off;

    // 1764 float4 stores per plane, ~7 iterations per thread.
    for (int t = threadIdx.x; t < QUADS; t += 256) {
        const int i  = t / ROWQ;            // output row
        const int c0 = (t - i * ROWQ) * 4;  // output col start

        int r = i + dy;
        r = r < 0 ? 0 : (r > H - 1 ? H - 1 : r);
        const float* rowp = &plane[r * W];

        const int j = c0 + dx;
        int j0 = j;     j0 = j0 < 0 ? 0 : (j0 > W - 1 ? W - 1 : j0);
        int j1 = j + 1; j1 = j1 < 0 ? 0 : (j1 > W - 1 ? W - 1 : j1);
        int j2 = j + 2; j2 = j2 < 0 ? 0 : (j2 > W - 1 ? W - 1 : j2);
        int j3 = j + 3; j3 = j3 < 0 ? 0 : (j3 > W - 1 ? W - 1 : j3);

        v4f v;
        v.x = rowp[j0];
        v.y = rowp[j1];
        v.z = rowp[j2];
        v.w = rowp[j3];

        // Write-once output: non-temporal b128 store, keep it out of L2.
        __builtin_nontemporal_store(v, (v4f*)(outp + i * W + c0));
    }
}

extern "C" void kernel_launch(void* const* d_in, const int* in_sizes, int n_in,
                              void* d_out, int out_size, void* d_ws, size_t ws_size,
                              hipStream_t stream) {
    const float* x     = (const float*)d_in[0];
    const int*   shift = (const int*)d_in[1];
    float*       out   = (float*)d_out;

    const int n_img = in_sizes[1] / 2;  // B*L (= 512 for the reference shapes)
    dim3 grid(C, n_img);
    drqv2_shift_tdm<<<grid, dim3(256), 0, stream>>>(x, shift, out);
}